// DenseEncodingLevel_23313082483302
// MI455X (gfx1250) — compile-verified
//
#include <hip/hip_runtime.h>

// Trilinear dense-grid encoding, 128^3 x 2 features, fp32.
// Bandwidth/gather-bound: table (16.8 MB) is L2-resident on MI455X (192 MB L2);
// coords/output are streamed with NT hints so they don't evict it. Corner
// pairs along z fetched as one b128 (cells z and z+1 are 16 contiguous bytes
// since F=2 is innermost). All addresses are (uniform SGPR base) + 32-bit
// unsigned BYTE offset so ISel selects the saddr addressing mode instead of
// building 64-bit VGPR address pairs per access.

typedef __attribute__((ext_vector_type(2))) float v2f;
typedef __attribute__((ext_vector_type(4))) float v4f;
// Reduced-alignment view of v4f: table gathers are only 8B-aligned
// (cell = 2 floats). gfx1250 VMEM handles dword-aligned b128.
typedef v4f v4f_a8 __attribute__((aligned(8)));

#define RM1 127

__device__ __forceinline__ const v4f_a8* tbl_at(const char* __restrict__ tb,
                                                unsigned byteOff) {
  return (const v4f_a8*)(tb + byteOff);   // base + zext(i32) -> saddr form
}

__device__ __forceinline__ v2f trilerp_one(const char* __restrict__ tb,
                                           float cx, float cy, float cz) {
  const float sx = cx * 127.0f;
  const float sy = cy * 127.0f;
  const float sz = cz * 127.0f;

  // Coords are >= 0, so int truncation == floor; clamp matches the
  // reference's clip(floor(x), 0, 127) for any edge case.
  int ix0 = (int)sx; ix0 = ix0 < 0 ? 0 : (ix0 > RM1 ? RM1 : ix0);
  int iy0 = (int)sy; iy0 = iy0 < 0 ? 0 : (iy0 > RM1 ? RM1 : iy0);
  int iz0 = (int)sz; iz0 = iz0 < 0 ? 0 : (iz0 > RM1 ? RM1 : iz0);

  const float fx = sx - (float)ix0;
  const float fy = sy - (float)iy0;
  const float fz = sz - (float)iz0;

  const unsigned ix1 = ix0 < RM1 ? (unsigned)ix0 + 1u : RM1;
  const unsigned iy1 = iy0 < RM1 ? (unsigned)iy0 + 1u : RM1;
  // Base z for the paired load: b128 covers cells izb and izb+1.
  const unsigned izb = iz0 < RM1 ? (unsigned)iz0 : (RM1 - 1);
  const bool z_clamped = (iz0 == RM1);   // then z0 == z1 == 127 (= hi half)

  // Byte offsets (cell = 8 bytes): ((x<<14)+(y<<7)+z) * 8  -> all < 2^25.
  const unsigned bx0 = (unsigned)ix0 << 17;   // * 128*128*8
  const unsigned bx1 = ix1 << 17;
  const unsigned by0 = (unsigned)iy0 << 10;   // * 128*8
  const unsigned by1 = iy1 << 10;
  const unsigned bz  = izb << 3;              // * 8

  // Each gather: 16 bytes = features of (.,.,izb) and (.,.,izb+1).
  const v4f g00 = *tbl_at(tb, bx0 + by0 + bz);
  const v4f g01 = *tbl_at(tb, bx0 + by1 + bz);
  const v4f g10 = *tbl_at(tb, bx1 + by0 + bz);
  const v4f g11 = *tbl_at(tb, bx1 + by1 + bz);

  const v2f h00 = {g00[2], g00[3]};
  const v2f h01 = {g01[2], g01[3]};
  const v2f h10 = {g10[2], g10[3]};
  const v2f h11 = {g11[2], g11[3]};
  const v2f l00 = z_clamped ? h00 : (v2f){g00[0], g00[1]};
  const v2f l01 = z_clamped ? h01 : (v2f){g01[0], g01[1]};
  const v2f l10 = z_clamped ? h10 : (v2f){g10[0], g10[1]};
  const v2f l11 = z_clamped ? h11 : (v2f){g11[0], g11[1]};

  // Blend z, then y, then x (contracted to v_fma).
  const v2f f00 = l00 + fz * (h00 - l00);
  const v2f f01 = l01 + fz * (h01 - l01);
  const v2f f10 = l10 + fz * (h10 - l10);
  const v2f f11 = l11 + fz * (h11 - l11);

  const v2f f0 = f00 + fy * (f01 - f00);
  const v2f f1 = f10 + fy * (f11 - f10);

  return f0 + fx * (f1 - f0);
}

// Hot kernel: one thread = one PAIR of points; fully straight-line.
__global__ __launch_bounds__(256) void trilerp128_pairs(
    const float* __restrict__ c0, const float* __restrict__ c1,
    const float* __restrict__ c2, const float* __restrict__ table,
    float* __restrict__ out, unsigned nPairs) {
  const unsigned t = blockIdx.x * 256u + threadIdx.x;
  if (t >= nPairs) return;
  const unsigned cOff = t << 3;    // byte offset of the coord pair (2 floats)

  // Paired, 8B-aligned, non-temporal streamed coord loads (b64, saddr form).
  const v2f a = __builtin_nontemporal_load((const v2f*)((const char*)c0 + cOff));
  const v2f b = __builtin_nontemporal_load((const v2f*)((const char*)c1 + cOff));
  const v2f c = __builtin_nontemporal_load((const v2f*)((const char*)c2 + cOff));

  // 8 independent b128 gathers in flight per thread -> deep MLP into L2.
  const char* tb = (const char*)table;
  const v2f r0 = trilerp_one(tb, a[0], b[0], c[0]);
  const v2f r1 = trilerp_one(tb, a[1], b[1], c[1]);

  const v4f res = {r0[0], r0[1], r1[0], r1[1]};
  __builtin_nontemporal_store(res, (v4f*)((char*)out + (t << 4)));  // 16B aligned
}

// Cold kernel: handles the single leftover point when n is odd (never for
// the harness's N = 4,194,304, but keeps kernel_launch correct for any n).
__global__ void trilerp128_tail(
    const float* __restrict__ c0, const float* __restrict__ c1,
    const float* __restrict__ c2, const float* __restrict__ table,
    float* __restrict__ out, unsigned p) {
  if (threadIdx.x == 0 && blockIdx.x == 0) {
    const v2f r = trilerp_one((const char*)table, c0[p], c1[p], c2[p]);
    out[2u * p + 0] = r[0];
    out[2u * p + 1] = r[1];
  }
}

extern "C" void kernel_launch(void* const* d_in, const int* in_sizes, int n_in,
                              void* d_out, int out_size, void* d_ws, size_t ws_size,
                              hipStream_t stream) {
  const float* c0    = (const float*)d_in[0];
  const float* c1    = (const float*)d_in[1];
  const float* c2    = (const float*)d_in[2];
  const float* table = (const float*)d_in[3];
  float* out         = (float*)d_out;

  const unsigned n      = (unsigned)in_sizes[0];   // 4,194,304 points
  const unsigned nPairs = n >> 1;                  // 2 points / thread
  if (nPairs > 0) {
    const unsigned blocks = (nPairs + 255u) / 256u;  // 8192 blocks of 8 wave32s
    trilerp128_pairs<<<blocks, 256, 0, stream>>>(c0, c1, c2, table, out, nPairs);
  }
  if (n & 1u) {
    trilerp128_tail<<<1, 1, 0, stream>>>(c0, c1, c2, table, out, n - 1u);
  }
}